// ODEINDLayer_79302276153901
// MI455X (gfx1250) — compile-verified
//
#include <hip/hip_runtime.h>
#include <math.h>

// ODEINDLayer for MI455X (gfx1250, wave32).
// B=128, S=128, E=4, D=1, O=4  ->  AtA is SPD banded, half-bandwidth 8.
// One workgroup (128 threads = 4 waves) per batch element:
//   Phase 1: Gram blocks C_s^T C_s via V_WMMA_F32_16X16X4_F32 (4 blocks/WMMA)
//   Phase 2: assemble 9-diagonal band + rhs entirely in LDS (~29 KB)
//   Phase 3: banded Cholesky (512 pivots, 36-lane trailing updates)
//   Phase 4: in-place banded forward/back solves, write x.

typedef __attribute__((ext_vector_type(2))) float v2f;
typedef __attribute__((ext_vector_type(8))) float v8f;

#define S_  128
#define E_  4
#define O_  4
#define N_  (S_ * O_)   // 512
#define BW  8           // half bandwidth
#define NK  (BW + 1)    // stored diagonals: band[n][k] = AtA[n][n-k]

__global__ __launch_bounds__(128)
void ode_band_chol_kernel(const float* __restrict__ coeff,   // (B,S,E,1,O)
                          const float* __restrict__ rhsE,    // (B,S,E)
                          const float* __restrict__ rinit,   // (B,2,1,2)
                          const float* __restrict__ steps,   // (B,S-1)
                          float* __restrict__ out)           // (B,S,1,O)
{
  __shared__ float band[N_][NK];     // 18 KB
  __shared__ float gram[S_][16];     //  8 KB  Gram blocks G_s[r][c]
  __shared__ float rhs[N_];          //  2 KB  beta -> y -> x (in place)
  __shared__ float sh_inv;

  const int b    = blockIdx.x;
  const int tid  = threadIdx.x;
  const int lane = tid & 31;
  const int wave = tid >> 5;

  // ---------------- Phase 1: Gram blocks via WMMA ----------------
  // Tile t covers steps 4t..4t+3. A = stacked C_s^T (16x4), B = stacked C_s
  // (4x16). With the ISA lane layouts for f32 A (lanes 0-15: K=0,1; lanes
  // 16-31: K=2,3) and B (VGPR v: lanes 0-15 K=v, 16-31 K=v+2), the per-lane
  // register contents of A and B coincide: coeff[b, s(blk), k, 0, j].
  {
    const int n16  = lane & 15;
    const int half = lane >> 4;
    const int blk  = n16 >> 2;    // which step inside the tile
    const int j    = n16 & 3;     // row/col index inside the 4x4 block
    const int e0   = half * 2;    // K pair held by this lane half

    __builtin_prefetch(&coeff[((size_t)b * S_) * E_ * O_], 0, 0);

    for (int t = wave; t < S_ / 4; t += 4) {
      const int    s  = t * 4 + blk;
      const size_t eb = ((size_t)b * S_ + s) * E_;
      v2f a;
      a.x = coeff[(eb + e0    ) * O_ + j];
      a.y = coeff[(eb + e0 + 1) * O_ + j];
      v8f cz = {};
      v8f dm = __builtin_amdgcn_wmma_f32_16x16x4_f32(
          false, a, false, a, (short)0, cz, false, false);
      // D[m][n]: lanes 0-15 hold M=v, lanes 16-31 hold M=v+8 (VGPR v).
      // Diagonal block of column-block blk lives in lanes 0-7 (blk 0,1)
      // and lanes 24-31 (blk 2,3).
      if ((blk >> 1) == half) {
        float* g = &gram[s][0];
        if (blk & 1) {
          g[0*4+j] = dm[4]; g[1*4+j] = dm[5]; g[2*4+j] = dm[6]; g[3*4+j] = dm[7];
        } else {
          g[0*4+j] = dm[0]; g[1*4+j] = dm[1]; g[2*4+j] = dm[2]; g[3*4+j] = dm[3];
        }
      }
    }
  }
  __syncthreads();

  // ---------------- Phase 2: band + rhs assembly (thread == step s) -------
  {
    const int s = tid;  // blockDim.x == S_
    const float SGN[4] = {1.f, -1.f, 1.f, -1.f};
    // fact[i][j] = 1/(j-i)! for j>=i else 0; fact[3][3] forced to 0.
    const float FCT[4][4] = {{1.f, 1.f, 0.5f, 1.f/6.f},
                             {0.f, 1.f, 1.f,  0.5f},
                             {0.f, 0.f, 1.f,  1.f},
                             {0.f, 0.f, 0.f,  0.f}};
    float FtF[4][4];
    #pragma unroll
    for (int r = 0; r < 4; ++r)
      #pragma unroll
      for (int c = 0; c < 4; ++c) {
        float acc = 0.f;
        #pragma unroll
        for (int i = 0; i < 4; ++i) acc += FCT[i][r] * FCT[i][c];
        FtF[r][c] = acc;
      }

    const float* st = steps + (size_t)b * (S_ - 1);
    const float hs  = (s <= S_ - 2) ? st[s]     : 0.f;
    const float hp  = (s >= 1)      ? st[s - 1] : 0.f;
    const float hpp = (s >= 2)      ? st[s - 2] : 0.f;
    const float hn  = (s <= S_ - 3) ? st[s + 1] : 0.f;

    float es[4], ep[4];
    es[0] = 1.f; ep[0] = 1.f;
    #pragma unroll
    for (int r = 1; r < 4; ++r) { es[r] = es[r-1]*hs; ep[r] = ep[r-1]*hp; }

    #pragma unroll
    for (int r = 0; r < 4; ++r)
      #pragma unroll
      for (int k = 0; k < NK; ++k)
        band[4*s + r][k] = 0.f;

    // diagonal block: Gram + et_ft_f_e(s) + et_ft_f_e(s-1)*sign_map
    #pragma unroll
    for (int r = 0; r < 4; ++r)
      #pragma unroll
      for (int c = 0; c <= r; ++c) {
        float v = gram[s][r*4 + c];
        if (s <= S_ - 2) v += es[r]*es[c]*FtF[r][c];
        if (s >= 1)      v += ep[r]*ep[c]*FtF[r][c]*SGN[r]*SGN[c];
        band[4*s + r][r - c] = v;
      }
    // et_e_diag = [1, h^2, h^4, 0] from both neighbors
    if (s <= S_ - 2) { band[4*s+0][0] += 1.f; band[4*s+1][0] += es[1]*es[1];
                       band[4*s+2][0] += es[2]*es[2]; }
    if (s >= 1)      { band[4*s+0][0] += 1.f; band[4*s+1][0] += ep[1]*ep[1];
                       band[4*s+2][0] += ep[2]*ep[2]; }
    // initial-value identity rows (init_idx = {0,1})
    if (s < 2) { band[4*s+0][0] += 1.f; band[4*s+1][0] += 1.f; }

    // steps2 = adjacent-step sums; powers 2*o-6=2, 2*o-5=3, 2*o-4=4
    const float s2a = hs + hn;    // steps[s]+steps[s+1]   (s <= S-3)
    const float s2b = hpp + hp;   // steps[s-2]+steps[s-1] (s >= 2)
    const float s2c = hp + hs;    // steps[s-1]+steps[s]   (1 <= s <= S-2)
    if (s <= S_ - 3) band[4*s+2][0] += s2a*s2a;
    if (s >= 2)      band[4*s+2][0] += s2b*s2b;
    if (s >= 1 && s <= S_ - 2) { const float q = s2c*s2c; band[4*s+3][0] += q*q; }
    if (s >= 2)      band[4*s+2][8] = -(s2b*s2b);            // sbd2 entry

    // sbd1 block: AtA[block s, block s-1]
    if (s >= 1) {
      float M1[4][4];
      #pragma unroll
      for (int r = 0; r < 4; ++r)
        #pragma unroll
        for (int c = 0; c < 4; ++c)
          M1[r][c] = ep[r]*ep[c] * (-(FCT[r][c] + FCT[c][r]*SGN[r]*SGN[c]));
      if (s <= S_ - 2) M1[3][2] += s2c*s2c*s2c;
      if (s >= 2)      M1[2][3] -= s2b*s2b*s2b;
      #pragma unroll
      for (int r = 0; r < 4; ++r)
        #pragma unroll
        for (int c = 0; c < 4; ++c)
          band[4*s + r][4 + r - c] = M1[r][c];
    }

    // beta = C^T rhs_eq (+ IV rhs)
    {
      const size_t cb = ((size_t)b * S_ + s) * E_;
      float ce[E_];
      #pragma unroll
      for (int e = 0; e < E_; ++e) ce[e] = rhsE[cb + e];
      #pragma unroll
      for (int r = 0; r < 4; ++r) {
        float acc = 0.f;
        #pragma unroll
        for (int e = 0; e < E_; ++e) acc += coeff[(cb + e)*O_ + r] * ce[e];
        if (s < 2 && r < 2) acc += rinit[((size_t)b*2 + s)*2 + r];
        rhs[4*s + r] = acc;
      }
    }
  }
  __syncthreads();

  // ---------------- Phase 3: banded Cholesky (right-looking) --------------
  // 36 trailing-update pairs (rr,cc), 1 <= cc <= rr <= 8.
  int urr = 0, ucc = 0;
  if (tid < 36) {
    int t = tid, r = 1;
    while (t >= r) { t -= r; ++r; }
    urr = r; ucc = t + 1;
  }
  for (int jj = 0; jj < N_; ++jj) {
    if (tid == 0) {
      const float d = sqrtf(band[jj][0]);
      band[jj][0] = d;
      sh_inv = 1.f / d;
    }
    __syncthreads();
    if (tid >= 1 && tid <= BW && jj + tid < N_)
      band[jj + tid][tid] *= sh_inv;                  // L[jj+k][jj]
    __syncthreads();
    if (tid < 36 && jj + urr < N_)                    // A[r][c] -= L[r][j]L[c][j]
      band[jj + urr][urr - ucc] -= band[jj + urr][urr] * band[jj + ucc][ucc];
    __syncthreads();
  }

  // ---------------- Phase 4: banded solves (in place) ---------------------
  if (tid == 0) {
    for (int n = 0; n < N_; ++n) {                    // L y = beta
      float t = rhs[n];
      const int kmax = n < BW ? n : BW;
      for (int k = 1; k <= kmax; ++k) t -= band[n][k] * rhs[n - k];
      rhs[n] = t / band[n][0];
    }
    for (int n = N_ - 1; n >= 0; --n) {               // L^T x = y
      float t = rhs[n];
      for (int k = 1; k <= BW && n + k < N_; ++k) t -= band[n + k][k] * rhs[n + k];
      rhs[n] = t / band[n][0];
    }
  }
  __syncthreads();
  for (int n = tid; n < N_; n += blockDim.x)
    out[(size_t)b * N_ + n] = rhs[n];
}

extern "C" void kernel_launch(void* const* d_in, const int* in_sizes, int n_in,
                              void* d_out, int out_size, void* d_ws, size_t ws_size,
                              hipStream_t stream) {
  (void)n_in; (void)out_size; (void)d_ws; (void)ws_size;
  const float* coeff = (const float*)d_in[0];   // (B,S,E,1,O) f32
  const float* rhsE  = (const float*)d_in[1];   // (B,S,E)     f32
  const float* rinit = (const float*)d_in[2];   // (B,2,1,2)   f32
  const float* steps = (const float*)d_in[3];   // (B,S-1)     f32
  float* out = (float*)d_out;                   // (B,S,1,O)   f32
  const int B = in_sizes[2] / 4;                // rhs_init has B*2*1*2 elems
  ode_band_chol_kernel<<<B, 128, 0, stream>>>(coeff, rhsE, rinit, steps, out);
}